// PatchAttacker_19258633356157
// MI455X (gfx1250) — compile-verified
//
#include <hip/hip_runtime.h>
#include <hip/hip_bf16.h>

// Problem constants (fixed by the reference).
#define Bn 32
#define Nn 6
#define Hn 512
#define Wn 512
#define Pp 512
#define SCALEf 0.5f
#define MIN_PHf 60.0f

#define AS1 __attribute__((address_space(1)))
#define AS3 __attribute__((address_space(3)))

// Single-pass patch paste:
//   out[b,y,x,:] = bilinear(patch, box_n)  for the LAST valid covering box n,
//                  else images[b,y,x,:]
// Image stream uses non-temporal hints; box table staged via CDNA5 async
// global->LDS DMA (ASYNCcnt) once per block.
__global__ __launch_bounds__(256) void paste_kernel(
    const float* __restrict__ images,
    const float* __restrict__ boxes,
    const float* __restrict__ patch,
    float* __restrict__ out)
{
    const int b   = blockIdx.y;
    const int tid = threadIdx.x;

    __shared__ float sbox[Nn * 4];

    // ---- Stage this image's 6 boxes (24 dwords) into LDS with the async path.
    if (tid < Nn * 4) {
        const float* gsrc = boxes + (size_t)b * (Nn * 4) + tid;
#if __has_builtin(__builtin_amdgcn_global_load_async_to_lds_b32)
        __builtin_amdgcn_global_load_async_to_lds_b32(
            (AS1 int*)gsrc, (AS3 int*)&sbox[tid], /*offset=*/0, /*cpol=*/0);
#else
        unsigned laddr = (unsigned)(size_t)(AS3 float*)&sbox[tid];
        unsigned long long gaddr = (unsigned long long)gsrc;
        asm volatile("global_load_async_to_lds_b32 %0, %1, off"
                     :: "v"(laddr), "v"(gaddr) : "memory");
#endif
    }
#if __has_builtin(__builtin_amdgcn_s_wait_asynccnt)
    __builtin_amdgcn_s_wait_asynccnt(0);
#else
    asm volatile("s_wait_asynccnt 0" ::: "memory");
#endif
    __syncthreads();

    // ---- Per-box patch geometry (uniform per block; fully unrolled, cheap).
    int   py0[Nn], px0[Nn], ph_i[Nn], pw_i[Nn];
    float sclY[Nn], sclX[Nn];
#pragma unroll
    for (int n = 0; n < Nn; ++n) {
        const float ymin = sbox[n * 4 + 0];
        const float xmin = sbox[n * 4 + 1];
        const float ymax = sbox[n * 4 + 2];
        const float xmax = sbox[n * 4 + 3];
        const float h  = ymax - ymin;
        const float w  = xmax - xmin;
        const float pw = h * SCALEf;
        const float ph = pw;                       // ASPECT == 1.0
        const float oy = ymin + h * 0.5f;
        const float ox = xmin + w * 0.5f;
        float yp = fmaxf(oy - ph * 0.5f, 0.0f);
        float xp = fmaxf(ox - pw * 0.5f, 0.0f);
        yp = (yp + ph > (float)Hn) ? ((float)Hn - ph) : yp;
        xp = (xp + pw > (float)Wn) ? ((float)Wn - pw) : xp;
        const bool valid = ph > MIN_PHf;
        const int ih = (int)ph;                    // astype(int32) truncation
        const int iw = (int)pw;
        py0[n]  = (int)yp;
        px0[n]  = (int)xp;
        ph_i[n] = valid ? ih : 0;                  // invalid -> never "inside"
        pw_i[n] = valid ? iw : 0;
        const float hf = fmaxf((float)ih, 1.0f);
        const float wf = fmaxf((float)iw, 1.0f);
        sclY[n] = (float)Pp / hf;
        sclX[n] = (float)Pp / wf;
    }

    // ---- One pixel per thread. H*W is a multiple of 256, no bounds check.
    const int pix = blockIdx.x * 256 + tid;
    const int y   = pix >> 9;          // W == 512
    const int x   = pix & (Wn - 1);

    const size_t base = ((size_t)b * (Hn * Wn) + pix) * 3;
    const float* ip = images + base;

    // Streamed (non-temporal) image read: don't pollute L2, keep patch resident.
    float c0 = __builtin_nontemporal_load(ip + 0);
    float c1 = __builtin_nontemporal_load(ip + 1);
    float c2 = __builtin_nontemporal_load(ip + 2);

    // Last-writer-wins: scan boxes from the last one back, take first cover.
#pragma unroll
    for (int n = Nn - 1; n >= 0; --n) {
        const int dy = y - py0[n];
        const int dx = x - px0[n];
        if ((unsigned)dy < (unsigned)ph_i[n] && (unsigned)dx < (unsigned)pw_i[n]) {
            float sy = ((float)dy + 0.5f) * sclY[n] - 0.5f;
            float sx = ((float)dx + 0.5f) * sclX[n] - 0.5f;
            sy = fminf(fmaxf(sy, 0.0f), (float)(Pp - 1));
            sx = fminf(fmaxf(sx, 0.0f), (float)(Pp - 1));
            const int yl = (int)sy;                // sy >= 0 -> trunc == floor
            const int xl = (int)sx;
            const float wy = sy - (float)yl;
            const float wx = sx - (float)xl;
            int yh = yl + 1; if (yh > Pp - 1) yh = Pp - 1;
            int xh = xl + 1; if (xh > Pp - 1) xh = Pp - 1;

            const float* p00 = patch + ((size_t)yl * Pp + xl) * 3;
            const float* p01 = patch + ((size_t)yl * Pp + xh) * 3;
            const float* p10 = patch + ((size_t)yh * Pp + xl) * 3;
            const float* p11 = patch + ((size_t)yh * Pp + xh) * 3;

            // Separable blend, same association as the reference (rows then cols).
            const float omwy = 1.0f - wy;
            const float omwx = 1.0f - wx;
#pragma unroll
            for (int c = 0; c < 3; ++c) {
                const float r0 = p00[c] * omwy + p10[c] * wy;
                const float r1 = p01[c] * omwy + p11[c] * wy;
                const float v  = r0 * omwx + r1 * wx;
                if (c == 0) c0 = v; else if (c == 1) c1 = v; else c2 = v;
            }
            break;
        }
    }

    float* op = out + base;
    __builtin_nontemporal_store(c0, op + 0);
    __builtin_nontemporal_store(c1, op + 1);
    __builtin_nontemporal_store(c2, op + 2);
}

extern "C" void kernel_launch(void* const* d_in, const int* in_sizes, int n_in,
                              void* d_out, int out_size, void* d_ws, size_t ws_size,
                              hipStream_t stream) {
    (void)in_sizes; (void)n_in; (void)out_size; (void)d_ws; (void)ws_size;
    const float* images = (const float*)d_in[0];   // [B,H,W,3]
    const float* boxes  = (const float*)d_in[1];   // [B,N,4]
    const float* patch  = (const float*)d_in[2];   // [P,P,3]
    float* out = (float*)d_out;                    // [B,H,W,3]

    dim3 grid((Hn * Wn) / 256, Bn, 1);
    paste_kernel<<<grid, 256, 0, stream>>>(images, boxes, patch, out);
}